// pointsEnhancement_2903397892256
// MI455X (gfx1250) — compile-verified
//
#include <hip/hip_runtime.h>

// Problem sizes (match reference)
constexpr int NB   = 8;       // batches
constexpr int NPTS = 16384;   // points per batch
constexpr int NG   = 2048;    // GROUP_NUMS
constexpr int KNN  = 64;      // GROUP_SIZE
constexpr int MT   = 16;      // centers per workgroup tile (one WMMA M dim)
constexpr int THREADS = 256;  // 8 waves of 32
constexpr int WAVES = THREADS / 32;
constexpr int NTILES = NPTS / 16;  // 1024 point tiles of 16

typedef float v2f __attribute__((ext_vector_type(2)));
typedef float v8f __attribute__((ext_vector_type(8)));
typedef unsigned int v4u __attribute__((ext_vector_type(4)));
typedef int v8i __attribute__((ext_vector_type(8)));
typedef int v4i __attribute__((ext_vector_type(4)));

// D = A(16x4) * B(4x16) via CDNA5 WMMA, f32 path.
// A lanes: 0-15 -> M=lane, K={0,1}; 16-31 -> M=lane-16, K={2,3}
// D lanes: VGPR v: lanes0-15 M=v, N=lane; lanes16-31 M=v+8, N=lane-16
__device__ __forceinline__ v8f wmma_dist16x16(v2f a, v2f b) {
  v8f c = {};
  c = __builtin_amdgcn_wmma_f32_16x16x4_f32(
      /*neg_a=*/false, a, /*neg_b=*/false, b,
      /*c_mod=*/(short)0, c, /*reuse_a=*/false, /*reuse_b=*/false);
  return c;
}

// ---------------------------------------------------------------------------
// Kernel 1: farthest point sampling. One block (1024 threads) per batch.
// ---------------------------------------------------------------------------
constexpr int FPS_THREADS = 1024;
constexpr int FPS_PPT = NPTS / FPS_THREADS;   // 16 points per thread

__global__ __launch_bounds__(FPS_THREADS)
void fps_kernel(const float* __restrict__ pts, float* __restrict__ centers) {
  __shared__ float sLast[3];
  __shared__ float wVal[32];
  __shared__ int   wIdx[32];
  const int b = blockIdx.x;
  const int t = threadIdx.x;
  const int lane = t & 31;
  const int wave = t >> 5;
  const float* P = pts + (size_t)b * NPTS * 3;

  float px[FPS_PPT], py[FPS_PPT], pz[FPS_PPT], dmin[FPS_PPT];
#pragma unroll
  for (int i = 0; i < FPS_PPT; ++i) {
    int n = t + FPS_THREADS * i;
    px[i] = P[(size_t)n * 3 + 0];
    py[i] = P[(size_t)n * 3 + 1];
    pz[i] = P[(size_t)n * 3 + 2];
    dmin[i] = 1.0e10f;
  }
  if (t == 0) { sLast[0] = P[0]; sLast[1] = P[1]; sLast[2] = P[2]; }

  for (int it = 0; it < NG; ++it) {
    __syncthreads();
    float lx = sLast[0], ly = sLast[1], lz = sLast[2];
    if (t == 0) {  // emit current 'last' as center (matches reference scan order)
      float* c = centers + ((size_t)b * NG + it) * 3;
      c[0] = lx; c[1] = ly; c[2] = lz;
    }
    float bv = -1.0f; int bi = 0x7fffffff;
#pragma unroll
    for (int i = 0; i < FPS_PPT; ++i) {
      float dx = px[i] - lx, dy = py[i] - ly, dz = pz[i] - lz;
      float d = dx * dx + dy * dy + dz * dz;
      dmin[i] = fminf(dmin[i], d);
      int n = t + FPS_THREADS * i;
      if (dmin[i] > bv || (dmin[i] == bv && n < bi)) { bv = dmin[i]; bi = n; }
    }
#pragma unroll
    for (int off = 16; off > 0; off >>= 1) {  // wave32 argmax
      float ov = __shfl_xor(bv, off, 32);
      int   oi = __shfl_xor(bi, off, 32);
      if (ov > bv || (ov == bv && oi < bi)) { bv = ov; bi = oi; }
    }
    if (lane == 0) { wVal[wave] = bv; wIdx[wave] = bi; }
    __syncthreads();
    if (wave == 0) {
      bv = wVal[lane]; bi = wIdx[lane];
#pragma unroll
      for (int off = 16; off > 0; off >>= 1) {
        float ov = __shfl_xor(bv, off, 32);
        int   oi = __shfl_xor(bi, off, 32);
        if (ov > bv || (ov == bv && oi < bi)) { bv = ov; bi = oi; }
      }
      if (lane == 0) {
        const float* w = P + (size_t)bi * 3;
        sLast[0] = w[0]; sLast[1] = w[1]; sLast[2] = w[2];
      }
    }
  }
}

// ---------------------------------------------------------------------------
// Kernel 2/3: k-NN (k=64) for a tile of 16 centers per workgroup.
// Points staged into 192KB LDS via Tensor Data Mover (one 24KB 1-D descriptor
// per wave), then 41 LDS-resident passes with v_wmma_f32_16x16x4_f32 tiles.
// ---------------------------------------------------------------------------
template <bool PASS2>
__global__ __launch_bounds__(THREADS)
void knn_kernel(const float* __restrict__ pts, const float* __restrict__ centers,
                float* __restrict__ outMeans, float* __restrict__ grouped,
                float* __restrict__ axisOut) {
  extern __shared__ float spts[];           // NPTS*3 floats = 192KB
  __shared__ float sLo[MT], sHi[MT], sMid[MT];
  __shared__ int   sCnt[MT], sSlot[MT];
  __shared__ float sSum[MT][3];
  __shared__ float sCen[MT][4];
  __shared__ float sGrp[MT][KNN][3];        // 12KB (used by PASS2)

  const int b = blockIdx.y;
  const int mBase = blockIdx.x * MT;
  const int t = threadIdx.x;
  const int lane = t & 31;
  const int wave = t >> 5;
  const int hv  = lane >> 4;                // 0: K={0,1}/M=v ; 1: K={2,3}/M=v+8
  const int col = lane & 15;

  if (t < MT) {
    const float* c = centers + ((size_t)b * NG + mBase + t) * 3;
    float cx = c[0], cy = c[1], cz = c[2];
    sCen[t][0] = cx; sCen[t][1] = cy; sCen[t][2] = cz;
    sCen[t][3] = cx * cx + cy * cy + cz * cz;
    sLo[t] = 0.0f; sHi[t] = 1.0e6f;
    sCnt[t] = 0; sSlot[t] = 0;
    sSum[t][0] = 0.0f; sSum[t][1] = 0.0f; sSum[t][2] = 0.0f;
  }

  // ---- Stage all points of this batch into LDS ----
#if __has_builtin(__builtin_amdgcn_tensor_load_to_lds) && \
    __has_builtin(__builtin_amdgcn_s_wait_tensorcnt)
  {
    // Each wave DMAs a contiguous 24KB slice: 6144 f32 elements.
    constexpr unsigned SLICE = (unsigned)(NPTS * 3 / WAVES);  // 6144
    unsigned ldsOff = (unsigned)(uintptr_t)&spts[0];          // low 32 bits = LDS byte offset
    unsigned long long ga =
        ((unsigned long long)(uintptr_t)(pts + ((size_t)b * NPTS * 3 + (size_t)wave * SLICE)))
        & 0x1FFFFFFFFFFFFFFULL;                               // 57-bit global address
    v4u g0;
    g0.x = 1u;                                   // count=1, user descriptor
    g0.y = ldsOff + (unsigned)wave * SLICE * 4u; // lds_addr (bytes)
    g0.z = (unsigned)(ga & 0xFFFFFFFFu);         // global_addr[31:0]
    g0.w = (unsigned)((ga >> 32) & 0x01FFFFFFu) | (2u << 30);  // addr[56:32] | type=2
    v8i g1;
    g1[0] = (int)(2u << 16);                     // workgroup_mask=0, data_size=2 (4B)
    g1[1] = (int)((SLICE & 0xFFFFu) << 16);      // tensor_dim0[15:0]
    g1[2] = (int)(((SLICE >> 16) & 0xFFFFu) | (1u << 16)); // tensor_dim0[31:16] | tensor_dim1=1
    g1[3] = (int)((SLICE & 0xFFFFu) << 16);      // tensor_dim1 hi=0 | tile_dim0=SLICE
    g1[4] = 0;                                   // tile_dim1=0 (unused), tile_dim2=0
    g1[5] = (int)SLICE;                          // tensor_dim0_stride[31:0]
    g1[6] = 0;                                   // stride hi / dim1_stride lo
    g1[7] = 0;
    v4i g2 = {0, 0, 0, 0};
    v4i g3 = {0, 0, 0, 0};
    v8i g4 = {0, 0, 0, 0, 0, 0, 0, 0};           // extra group (clang-23 6-arg form)
    __builtin_amdgcn_tensor_load_to_lds(g0, g1, g2, g3, g4, 0);
    __builtin_amdgcn_s_wait_tensorcnt(0);
  }
#else
  for (int i = t; i < NPTS * 3; i += THREADS)
    spts[i] = pts[(size_t)b * NPTS * 3 + i];
#endif
  __syncthreads();

  // Per-lane A operand: center row = (-2cx,-2cy,-2cz,|c|^2), branchless select
  const float cAx = sCen[col][0], cAy = sCen[col][1];
  const float cAz = sCen[col][2], cA2 = sCen[col][3];
  v2f a;
  a.x = hv ? (-2.0f * cAz) : (-2.0f * cAx);
  a.y = hv ? cA2           : (-2.0f * cAy);

  // ---- bisection on distance threshold (40 LDS-resident count passes) ----
  for (int it = 0; it < 40; ++it) {
    if (t < MT) { sMid[t] = 0.5f * (sLo[t] + sHi[t]); sCnt[t] = 0; }
    __syncthreads();
    float midr[8];
#pragma unroll
    for (int v = 0; v < 8; ++v) midr[v] = sMid[v + 8 * hv];
    int cnt[8] = {0, 0, 0, 0, 0, 0, 0, 0};
    for (int tile = wave; tile < NTILES; tile += WAVES) {
      int n = tile * 16 + col;
      float x = spts[n * 3 + 0], y = spts[n * 3 + 1], z = spts[n * 3 + 2];
      float p2 = x * x + y * y + z * z;
      v2f bb;
      bb.x = hv ? z : x;       // v_cndmask, no divergence
      bb.y = hv ? 1.0f : y;
      v8f d = wmma_dist16x16(a, bb);
#pragma unroll
      for (int v = 0; v < 8; ++v) {
        float dv = d[v] + p2;  // + |p|^2 for this lane's column
        cnt[v] += (dv <= midr[v]) ? 1 : 0;
      }
    }
#pragma unroll
    for (int off = 1; off < 16; off <<= 1) {   // reduce across 16-lane group
#pragma unroll
      for (int v = 0; v < 8; ++v) cnt[v] += __shfl_xor(cnt[v], off, 32);
    }
    if (col == 0) {
#pragma unroll
      for (int v = 0; v < 8; ++v) atomicAdd(&sCnt[v + 8 * hv], cnt[v]);
    }
    __syncthreads();
    if (t < MT) { if (sCnt[t] >= KNN) sHi[t] = sMid[t]; else sLo[t] = sMid[t]; }
    __syncthreads();
  }

  // ---- selection pass at threshold sHi (guaranteed >= 64 candidates) ----
  float thr[8];
#pragma unroll
  for (int v = 0; v < 8; ++v) thr[v] = sHi[v + 8 * hv];
  for (int tile = wave; tile < NTILES; tile += WAVES) {
    int n = tile * 16 + col;
    float x = spts[n * 3 + 0], y = spts[n * 3 + 1], z = spts[n * 3 + 2];
    float p2 = x * x + y * y + z * z;
    v2f bb;
    bb.x = hv ? z : x;
    bb.y = hv ? 1.0f : y;
    v8f d = wmma_dist16x16(a, bb);
#pragma unroll
    for (int v = 0; v < 8; ++v) {
      float dv = d[v] + p2;
      if (dv <= thr[v]) {
        int mrow = v + 8 * hv;
        int slot = atomicAdd(&sSlot[mrow], 1);
        if (slot < KNN) {
          atomicAdd(&sSum[mrow][0], x);
          atomicAdd(&sSum[mrow][1], y);
          atomicAdd(&sSum[mrow][2], z);
          if (PASS2) {
            sGrp[mrow][slot][0] = x;
            sGrp[mrow][slot][1] = y;
            sGrp[mrow][slot][2] = z;
            size_t g = (((size_t)b * NG + mBase + mrow) * KNN + slot) * 3;
            grouped[g + 0] = x; grouped[g + 1] = y; grouped[g + 2] = z;
          }
        }
      }
    }
  }
  __syncthreads();

  if (!PASS2) {
    if (t < MT) {
      size_t o = ((size_t)b * NG + mBase + t) * 3;
      const float s = 1.0f / (float)KNN;
      outMeans[o + 0] = sSum[t][0] * s;
      outMeans[o + 1] = sSum[t][1] * s;
      outMeans[o + 2] = sSum[t][2] * s;
    }
  } else {
    if (t < MT) {
      const float s = 1.0f / (float)KNN;
      float mx = sSum[t][0] * s, my = sSum[t][1] * s, mz = sSum[t][2] * s;
      float c00 = 0, c01 = 0, c02 = 0, c11 = 0, c12 = 0, c22 = 0;
      for (int k = 0; k < KNN; ++k) {
        float dx = sGrp[t][k][0] - mx;
        float dy = sGrp[t][k][1] - my;
        float dz = sGrp[t][k][2] - mz;
        c00 += dx * dx; c01 += dx * dy; c02 += dx * dz;
        c11 += dy * dy; c12 += dy * dz; c22 += dz * dz;
      }
      c00 *= s; c01 *= s; c02 *= s; c11 *= s; c12 *= s; c22 *= s;
      // dominant eigenvector (PSD 3x3) via power iteration
      float vx = 1.0f, vy = 1.0f, vz = 1.0f;
      for (int pi = 0; pi < 64; ++pi) {
        float wx = c00 * vx + c01 * vy + c02 * vz;
        float wy = c01 * vx + c11 * vy + c12 * vz;
        float wz = c02 * vx + c12 * vy + c22 * vz;
        float nrm = sqrtf(wx * wx + wy * wy + wz * wz);
        if (nrm > 1e-20f) { float inv = 1.0f / nrm; vx = wx * inv; vy = wy * inv; vz = wz * inv; }
      }
      size_t o = ((size_t)b * NG + mBase + t) * 3;
      axisOut[o + 0] = vx; axisOut[o + 1] = vy; axisOut[o + 2] = vz;
    }
  }
}

// ---------------------------------------------------------------------------
// Kernel 4: assemble output: linspace-subsample of [grouped | axis] rows.
// ---------------------------------------------------------------------------
__global__ __launch_bounds__(256)
void out_kernel(const float* __restrict__ grouped, const float* __restrict__ axis,
                float* __restrict__ out) {
  int gid = blockIdx.x * blockDim.x + threadIdx.x;
  if (gid >= NB * NPTS) return;
  int b = gid / NPTS, r = gid % NPTS;
  int idx = (int)((double)r * (double)(NG * KNN - 1) / (double)(NPTS - 1));
  int g = idx >> 6, k = idx & (KNN - 1);
  const float* gp = grouped + (((size_t)b * NG + g) * KNN + k) * 3;
  const float* ax = axis + ((size_t)b * NG + g) * 3;
  float* o = out + (size_t)gid * 6;
  o[0] = gp[0]; o[1] = gp[1]; o[2] = gp[2];
  o[3] = ax[0]; o[4] = ax[1]; o[5] = ax[2];
}

// ---------------------------------------------------------------------------
extern "C" void kernel_launch(void* const* d_in, const int* in_sizes, int n_in,
                              void* d_out, int out_size, void* d_ws, size_t ws_size,
                              hipStream_t stream) {
  (void)in_sizes; (void)n_in; (void)out_size; (void)ws_size;
  const float* pts = (const float*)d_in[0];
  float* ws = (float*)d_ws;
  float* centers = ws;                                   // NB*NG*3
  float* means   = ws + (size_t)NB * NG * 3;             // NB*NG*3
  float* axisv   = ws + (size_t)2 * NB * NG * 3;         // NB*NG*3
  float* grouped = ws + (size_t)3 * NB * NG * 3;         // NB*NG*KNN*3 (12MB)
  float* out = (float*)d_out;

  fps_kernel<<<dim3(NB), FPS_THREADS, 0, stream>>>(pts, centers);

  size_t ldsBytes = (size_t)NPTS * 3 * sizeof(float);    // 192KB of the 320KB WGP LDS
  dim3 knnGrid(NG / MT, NB);
  knn_kernel<false><<<knnGrid, THREADS, ldsBytes, stream>>>(pts, centers, means,
                                                            nullptr, nullptr);
  knn_kernel<true><<<knnGrid, THREADS, ldsBytes, stream>>>(pts, means, nullptr,
                                                           grouped, axisv);

  out_kernel<<<dim3((NB * NPTS + 255) / 256), 256, 0, stream>>>(grouped, axisv, out);
}